// DGCRL_69501160784007
// MI455X (gfx1250) — compile-verified
//
#include <hip/hip_runtime.h>

// ---------------------------------------------------------------------------
// DGCRL forward (2x GAT encoder + MLP head) for MI455X / gfx1250.
//
// Bandwidth-bound problem (~5 GB of L2-resident edge traffic vs ~8.2 GFLOP
// of GEMM), so GEMMs use full-precision V_WMMA_F32_16X16X4_F32: zero perf
// cost at 23.3 TB/s HBM / 192 MB L2, full fp32 numerics.
// ---------------------------------------------------------------------------

#define NN   100000
#define EE   1600000
#define FIN  128
#define FHID 64
#define NEGS 0.2f
#define LNEPS 1e-5f

typedef float v2f __attribute__((ext_vector_type(2)));
typedef float v8f __attribute__((ext_vector_type(8)));

// ---------------------------------------------------------------------------
// float atomic max via int/uint views (portable, no header dependence)
// ---------------------------------------------------------------------------
__device__ __forceinline__ void atomicMaxFloat(float* addr, float value) {
  if (value >= 0.0f) atomicMax((int*)addr, __float_as_int(value));
  else               atomicMin((unsigned int*)addr, __float_as_uint(value));
}

// ---------------------------------------------------------------------------
// GEMM: Out[n,FO] = X[n,fi] @ W[fi,FO]  (+bias, +PReLU) with f32 WMMA.
// One wave computes a 16xFO strip; 8 waves (256 thr) per block -> 128 rows.
// A 16x4 f32 fragment: lane<16 holds K={k,k+1}, lane>=16 holds K={k+2,k+3},
// row = lane&15. B 4x16: same K split, col = lane&15. C: vgpr v holds
// row = v + 8*(lane>=16), col = lane&15.   N is a multiple of 16.
// ---------------------------------------------------------------------------
template <int FO>
__global__ void gat_gemm_wmma(const float* __restrict__ X,
                              const float* __restrict__ Wm,
                              const float* __restrict__ bias,
                              const float* __restrict__ prelu,
                              float* __restrict__ Out,
                              int n, int fi) {
  constexpr int NT = FO / 16;
  const int lane = threadIdx.x & 31;
  const int wave = threadIdx.x >> 5;
  const int rowBase = blockIdx.x * 128 + wave * 16;
  if (rowBase >= n) return;                 // whole-wave exit: EXEC stays all-1s
  const int r  = lane & 15;
  const int kh = (lane >> 4) << 1;          // 0 or 2
  const int row = rowBase + r;

  v8f acc[NT];
#pragma unroll
  for (int t = 0; t < NT; ++t)
#pragma unroll
    for (int v = 0; v < 8; ++v) acc[t][v] = 0.0f;

  const float* xrow = X + (size_t)row * fi;
  for (int k = 0; k < fi; k += 4) {
    const int k0 = k + kh;
    v2f a;
    a.x = xrow[k0];
    a.y = xrow[k0 + 1];
    const float* w0 = Wm + (size_t)k0 * FO;
#pragma unroll
    for (int t = 0; t < NT; ++t) {
      v2f b;
      b.x = w0[t * 16 + r];
      b.y = w0[FO + t * 16 + r];
      acc[t] = __builtin_amdgcn_wmma_f32_16x16x4_f32(
          /*neg_a=*/false, a, /*neg_b=*/false, b,
          /*c_mod=*/(short)0, acc[t], /*reuse_a=*/false, /*reuse_b=*/false);
    }
  }

  const int rhalf = (lane >> 4) << 3;       // 0 or 8
#pragma unroll
  for (int t = 0; t < NT; ++t) {
#pragma unroll
    for (int v = 0; v < 8; ++v) {
      const int orow = rowBase + rhalf + v;
      const int ocol = t * 16 + r;
      float val = acc[t][v];
      if (bias)  val += bias[ocol];
      if (prelu) { float pa = prelu[0]; val = (val >= 0.0f) ? val : pa * val; }
      Out[(size_t)orow * FO + ocol] = val;
    }
  }
}

// ---------------------------------------------------------------------------
// Per-node attention scores: ssrc[i] = h[i].a_s ; sdst[i] = h[i].a_d  (F=64)
// One wave per node, shuffle reduction.
// ---------------------------------------------------------------------------
__global__ void node_scores(const float* __restrict__ H,
                            const float* __restrict__ a_s,
                            const float* __restrict__ a_d,
                            float* __restrict__ ssrc,
                            float* __restrict__ sdst, int n) {
  const int node = (int)((blockIdx.x * blockDim.x + threadIdx.x) >> 5);
  const int lane = threadIdx.x & 31;
  if (node >= n) return;
  const float* hrow = H + (size_t)node * 64;
  const float h0 = hrow[lane], h1 = hrow[lane + 32];
  float ps = h0 * a_s[lane] + h1 * a_s[lane + 32];
  float pd = h0 * a_d[lane] + h1 * a_d[lane + 32];
#pragma unroll
  for (int off = 16; off; off >>= 1) {
    ps += __shfl_xor(ps, off, 32);
    pd += __shfl_xor(pd, off, 32);
  }
  if (lane == 0) { ssrc[node] = ps; sdst[node] = pd; }
}

// ---------------------------------------------------------------------------
// Per-layer init: agg = broadcast(bias) (folds "+b" into the aggregation),
// m = -inf, z = 0, LN partial sums = 0.
// ---------------------------------------------------------------------------
__global__ void layer_init(float* __restrict__ agg, const float* __restrict__ bias,
                           float* __restrict__ m, float* __restrict__ z,
                           float* __restrict__ lnsums, int n) {
  const int idx = blockIdx.x * blockDim.x + threadIdx.x;
  const int total = n * 64;
  if (idx >= total) return;
  agg[idx] = bias[idx & 63];
  if (idx < n) { m[idx] = -__builtin_inff(); z[idx] = 0.0f; }
  if (idx < 2) lnsums[idx] = 0.0f;
}

// ---------------------------------------------------------------------------
// Edge softmax passes. Edges [0,E) come from the index arrays; edges
// [E, E+n) are the implicit self-loops (src = dst = e - E).
// ---------------------------------------------------------------------------
__global__ void edge_logits(const int* __restrict__ src, const int* __restrict__ dst,
                            const float* __restrict__ ssrc, const float* __restrict__ sdst,
                            float* __restrict__ le, float* __restrict__ m, int E, int n) {
  const int idx = blockIdx.x * blockDim.x + threadIdx.x;
  if (idx >= E + n) return;
  int s, d;
  if (idx < E) { s = src[idx]; d = dst[idx]; } else { s = d = idx - E; }
  float v = ssrc[s] + sdst[d];
  v = (v >= 0.0f) ? v : NEGS * v;           // LeakyReLU(0.2)
  le[idx] = v;
  atomicMaxFloat(m + d, v);
}

__global__ void edge_exp(const int* __restrict__ dst, float* __restrict__ le,
                         const float* __restrict__ m, float* __restrict__ z,
                         int E, int n) {
  const int idx = blockIdx.x * blockDim.x + threadIdx.x;
  if (idx >= E + n) return;
  const int d = (idx < E) ? dst[idx] : (idx - E);
  const float w = __expf(le[idx] - m[d]);
  le[idx] = w;
  atomicAdd(z + d, w);
}

__global__ void edge_alpha(const int* __restrict__ dst, float* __restrict__ le,
                           const float* __restrict__ z, int E, int n) {
  const int idx = blockIdx.x * blockDim.x + threadIdx.x;
  if (idx >= E + n) return;
  const int d = (idx < E) ? dst[idx] : (idx - E);
  le[idx] = le[idx] / z[d];
}

// 16 threads per edge, float4 per thread: coalesced 256B gather of h[src],
// fp32 atomic scatter into agg[dst]. All of h/agg is L2-resident (192 MB).
__global__ void edge_scatter(const int* __restrict__ src, const int* __restrict__ dst,
                             const float* __restrict__ le, const float* __restrict__ H,
                             float* __restrict__ agg, int E, int n) {
  const int idx = blockIdx.x * blockDim.x + threadIdx.x;
  const int total = (E + n) * 16;
  if (idx >= total) return;
  const int e  = idx >> 4;
  const int fg = idx & 15;
  int s, d;
  if (e < E) { s = src[e]; d = dst[e]; } else { s = d = e - E; }
  const float alpha = le[e];
  const float4 hv = *reinterpret_cast<const float4*>(H + (size_t)s * 64 + fg * 4);
  float* o = agg + (size_t)d * 64 + fg * 4;
  atomicAdd(o + 0, alpha * hv.x);
  atomicAdd(o + 1, alpha * hv.y);
  atomicAdd(o + 2, alpha * hv.z);
  atomicAdd(o + 3, alpha * hv.w);
}

// ---------------------------------------------------------------------------
// Graph-LayerNorm: global sum / sumsq reduction, then normalize + affine +
// PReLU (writes to `out`, which may alias the input or be a d_out slice).
// ---------------------------------------------------------------------------
__global__ void ln_reduce(const float* __restrict__ x, float* __restrict__ lnsums,
                          int total) {
  float s = 0.0f, s2 = 0.0f;
  for (int i = blockIdx.x * blockDim.x + threadIdx.x; i < total;
       i += gridDim.x * blockDim.x) {
    const float v = x[i];
    s += v; s2 += v * v;
  }
#pragma unroll
  for (int off = 16; off; off >>= 1) {
    s  += __shfl_xor(s,  off, 32);
    s2 += __shfl_xor(s2, off, 32);
  }
  __shared__ float sh[16];
  const int lane = threadIdx.x & 31, wid = threadIdx.x >> 5;
  if (lane == 0) { sh[wid] = s; sh[8 + wid] = s2; }
  __syncthreads();
  if (threadIdx.x == 0) {
    float ts = 0.0f, ts2 = 0.0f;
    for (int w = 0; w < (int)(blockDim.x >> 5); ++w) { ts += sh[w]; ts2 += sh[8 + w]; }
    atomicAdd(&lnsums[0], ts);
    atomicAdd(&lnsums[1], ts2);
  }
}

__global__ void ln_norm_prelu(const float* __restrict__ xin,
                              const float* __restrict__ lnsums,
                              const float* __restrict__ g,
                              const float* __restrict__ be,
                              const float* __restrict__ p,
                              float* __restrict__ out, int total) {
  const int idx = blockIdx.x * blockDim.x + threadIdx.x;
  if (idx >= total) return;
  const float inv = 1.0f / (float)total;
  const float mu  = lnsums[0] * inv;
  const float var = lnsums[1] * inv - mu * mu;
  const float sc  = rsqrtf(var + LNEPS);
  const int col = idx & 63;
  float v = (xin[idx] - mu) * sc * g[col] + be[col];
  const float pa = p[0];
  out[idx] = (v >= 0.0f) ? v : pa * v;
}

// ---------------------------------------------------------------------------
// Host-side orchestration
// ---------------------------------------------------------------------------
struct LayerP {
  const float *W, *as_, *ad_, *b, *g, *be, *p;
};

static void run_gat_layer(const float* x, int fi,
                          const int* src, const int* dst,
                          const LayerP& L,
                          float* h, float* agg, float* ssrc, float* sdst,
                          float* m, float* z, float* le, float* lnsums,
                          float* outbuf, hipStream_t stream) {
  const int n = NN, E = EE;
  const int ET = E + n;

  gat_gemm_wmma<64><<<(n + 127) / 128, 256, 0, stream>>>(
      x, L.W, nullptr, nullptr, h, n, fi);
  node_scores<<<(n + 7) / 8, 256, 0, stream>>>(h, L.as_, L.ad_, ssrc, sdst, n);
  layer_init<<<(n * 64 + 255) / 256, 256, 0, stream>>>(agg, L.b, m, z, lnsums, n);
  edge_logits<<<(ET + 255) / 256, 256, 0, stream>>>(src, dst, ssrc, sdst, le, m, E, n);
  edge_exp<<<(ET + 255) / 256, 256, 0, stream>>>(dst, le, m, z, E, n);
  edge_alpha<<<(ET + 255) / 256, 256, 0, stream>>>(dst, le, z, E, n);
  edge_scatter<<<(ET * 16 + 255) / 256, 256, 0, stream>>>(src, dst, le, h, agg, E, n);
  ln_reduce<<<1024, 256, 0, stream>>>(agg, lnsums, n * 64);
  ln_norm_prelu<<<(n * 64 + 255) / 256, 256, 0, stream>>>(
      agg, lnsums, L.g, L.be, L.p, outbuf, n * 64);
}

extern "C" void kernel_launch(void* const* d_in, const int* in_sizes, int n_in,
                              void* d_out, int out_size, void* d_ws, size_t ws_size,
                              hipStream_t stream) {
  (void)in_sizes; (void)n_in; (void)out_size; (void)ws_size;
  const int n = NN, E = EE;

  const float* x_q  = (const float*)d_in[0];
  const float* x_k  = (const float*)d_in[1];
  const int*   ei_q = (const int*)d_in[2];
  const int*   ei_k = (const int*)d_in[3];

  auto layer = [&](int base) -> LayerP {
    return LayerP{(const float*)d_in[base + 0], (const float*)d_in[base + 1],
                  (const float*)d_in[base + 2], (const float*)d_in[base + 3],
                  (const float*)d_in[base + 4], (const float*)d_in[base + 5],
                  (const float*)d_in[base + 6]};
  };
  const LayerP q1 = layer(4),  q2 = layer(11);
  const LayerP k1 = layer(18), k2 = layer(25);
  const float* m_W1 = (const float*)d_in[32];
  const float* m_b1 = (const float*)d_in[33];
  const float* m_p  = (const float*)d_in[34];
  const float* m_W2 = (const float*)d_in[35];
  const float* m_b2 = (const float*)d_in[36];

  // Workspace carving (256B aligned): h[N*128], agg[N*64], ssrc/sdst/m/z[N],
  // le[E+N], lnsums[2]  (~85 MB total)
  char* base = (char*)d_ws;
  size_t off = 0;
  auto carve = [&](size_t bytes) -> float* {
    float* p = (float*)(base + off);
    off = (off + bytes + 255) & ~(size_t)255;
    return p;
  };
  float* h      = carve((size_t)n * 128 * 4);
  float* agg    = carve((size_t)n * 64 * 4);
  float* ssrc   = carve((size_t)n * 4);
  float* sdst   = carve((size_t)n * 4);
  float* mbuf   = carve((size_t)n * 4);
  float* zbuf   = carve((size_t)n * 4);
  float* le     = carve((size_t)(E + n) * 4);
  float* lnsums = carve(256);

  float* out_q = (float*)d_out;
  float* out_k = (float*)d_out + (size_t)n * 64;

  // ---- Q encoder ----
  run_gat_layer(x_q, FIN,  ei_q, ei_q + E, q1, h, agg, ssrc, sdst, mbuf, zbuf, le, lnsums, agg, stream);
  run_gat_layer(agg, FHID, ei_q, ei_q + E, q2, h, agg, ssrc, sdst, mbuf, zbuf, le, lnsums, agg, stream);
  // ---- Q MLP head: out_q = PReLU(agg@W1 + b1) @ W2 + b2 ----
  gat_gemm_wmma<128><<<(n + 127) / 128, 256, 0, stream>>>(agg, m_W1, m_b1, m_p, h, n, 64);
  gat_gemm_wmma<64><<<(n + 127) / 128, 256, 0, stream>>>(h, m_W2, m_b2, nullptr, out_q, n, 128);
  // ---- K encoder (final layer writes straight into d_out second half) ----
  run_gat_layer(x_k, FIN,  ei_k, ei_k + E, k1, h, agg, ssrc, sdst, mbuf, zbuf, le, lnsums, agg, stream);
  run_gat_layer(agg, FHID, ei_k, ei_k + E, k2, h, agg, ssrc, sdst, mbuf, zbuf, le, lnsums, out_k, stream);
}